// Fit_532575945089
// MI455X (gfx1250) — compile-verified
//
#include <hip/hip_runtime.h>

typedef __attribute__((ext_vector_type(2))) float v2f;
typedef __attribute__((ext_vector_type(8))) float v8f;

#define N_PTS   512
#define THREADS 256
#define WAVES   8          // 256 threads / wave32
#define XSTRIDE 10         // 9 components + 1 pad (conflict-free, allows m up to 15 reads)

// Serial cyclic Jacobi eigen-decomposition of symmetric n x n A (row major).
// On exit: diag(A) = eigenvalues, V columns = eigenvectors.
__device__ inline void jacobi_sym(float* A, float* V, int n, int sweeps) {
    for (int i = 0; i < n * n; ++i) V[i] = 0.0f;
    for (int i = 0; i < n; ++i) V[i * n + i] = 1.0f;
    for (int s = 0; s < sweeps; ++s) {
        for (int p = 0; p < n - 1; ++p) {
            for (int q = p + 1; q < n; ++q) {
                float apq = A[p * n + q];
                if (fabsf(apq) < 1e-25f) continue;
                float app = A[p * n + p], aqq = A[q * n + q];
                float tau = (aqq - app) / (2.0f * apq);
                float t = (tau >= 0.0f ? 1.0f : -1.0f) /
                          (fabsf(tau) + sqrtf(tau * tau + 1.0f));
                float c = 1.0f / sqrtf(t * t + 1.0f);
                float sn = t * c;
                for (int k = 0; k < n; ++k) {            // A <- A * J
                    float akp = A[k * n + p], akq = A[k * n + q];
                    A[k * n + p] = c * akp - sn * akq;
                    A[k * n + q] = sn * akp + c * akq;
                }
                for (int k = 0; k < n; ++k) {            // A <- J^T * A
                    float apk = A[p * n + k], aqk = A[q * n + k];
                    A[p * n + k] = c * apk - sn * aqk;
                    A[q * n + k] = sn * apk + c * aqk;
                }
                for (int k = 0; k < n; ++k) {            // V <- V * J
                    float vkp = V[k * n + p], vkq = V[k * n + q];
                    V[k * n + p] = c * vkp - sn * vkq;
                    V[k * n + q] = sn * vkp + c * vkq;
                }
            }
        }
    }
}

__global__ __launch_bounds__(THREADS)
void fit_fundamental_kernel(const float* __restrict__ pts1,
                            const float* __restrict__ pts2,
                            const float* __restrict__ wts,
                            float* __restrict__ outF,   // [B,3,3]
                            float* __restrict__ outR)   // [B,N]
{
    __shared__ float Xs[N_PTS * XSTRIDE + 16];  // weighted normalized DLT rows
    __shared__ float sPart[WAVES * 256];        // WMMA partials / reduction scratch
    __shared__ float sA[81];                    // 9x9 Gram
    __shared__ float sV[81];
    __shared__ float sM3[9], sV3[9];
    __shared__ float sFvec[9];
    __shared__ float sSc[8];                    // s1, c1x, c1y, s2, c2x, c2y

    const int tid = threadIdx.x;
    const int b   = blockIdx.x;
    const float* P1 = pts1 + (size_t)b * N_PTS * 3;
    const float* P2 = pts2 + (size_t)b * N_PTS * 3;
    const float* W  = wts  + (size_t)b * N_PTS;

    // ---- load 2 points per thread ----
    float x1[2], y1[2], x2[2], y2[2], ww[2];
#pragma unroll
    for (int j = 0; j < 2; ++j) {
        int i = tid + j * THREADS;
        x1[j] = P1[i * 3 + 0];  y1[j] = P1[i * 3 + 1];
        x2[j] = P2[i * 3 + 0];  y2[j] = P2[i * 3 + 1];
        ww[j] = W[i];
    }

    // ---- centroid reduction (float4: sx1, sy1, sx2, sy2) ----
    float4* red4 = (float4*)sPart;
    red4[tid] = make_float4(x1[0] + x1[1], y1[0] + y1[1],
                            x2[0] + x2[1], y2[0] + y2[1]);
    __syncthreads();
    for (int s = THREADS / 2; s > 0; s >>= 1) {
        if (tid < s) {
            float4 a = red4[tid], c = red4[tid + s];
            red4[tid] = make_float4(a.x + c.x, a.y + c.y, a.z + c.z, a.w + c.w);
        }
        __syncthreads();
    }
    if (tid == 0) {
        const float inv = 1.0f / (float)N_PTS;
        sSc[1] = red4[0].x * inv;  sSc[2] = red4[0].y * inv;
        sSc[4] = red4[0].z * inv;  sSc[5] = red4[0].w * inv;
    }
    __syncthreads();
    const float c1x = sSc[1], c1y = sSc[2], c2x = sSc[4], c2y = sSc[5];

    // ---- mean radial distance reduction (float2) ----
    float2* red2 = (float2*)sPart;
    float d1 = 0.0f, d2 = 0.0f;
#pragma unroll
    for (int j = 0; j < 2; ++j) {
        float dx = x1[j] - c1x, dy = y1[j] - c1y;
        d1 += sqrtf(dx * dx + dy * dy);
        dx = x2[j] - c2x;  dy = y2[j] - c2y;
        d2 += sqrtf(dx * dx + dy * dy);
    }
    __syncthreads();
    red2[tid] = make_float2(d1, d2);
    __syncthreads();
    for (int s = THREADS / 2; s > 0; s >>= 1) {
        if (tid < s) {
            float2 a = red2[tid], c = red2[tid + s];
            red2[tid] = make_float2(a.x + c.x, a.y + c.y);
        }
        __syncthreads();
    }
    if (tid == 0) {
        sSc[0] = 1.4142f * (float)N_PTS / red2[0].x;   // scale = SQRT2 / meandist
        sSc[3] = 1.4142f * (float)N_PTS / red2[0].y;
    }
    __syncthreads();
    const float s1 = sSc[0], s2 = sSc[3];

    // ---- build weighted, row-normalized DLT rows into LDS ----
#pragma unroll
    for (int j = 0; j < 2; ++j) {
        int i = tid + j * THREADS;
        float ax = s1 * (x1[j] - c1x), ay = s1 * (y1[j] - c1y);
        float bx = s2 * (x2[j] - c2x), by = s2 * (y2[j] - c2y);
        float v[9] = { bx * ax, bx * ay, bx,
                       by * ax, by * ay, by,
                       ax,      ay,      1.0f };
        float nrm = 0.0f;
#pragma unroll
        for (int c = 0; c < 9; ++c) nrm += v[c] * v[c];
        nrm = sqrtf(nrm);
        float sc = ww[j] / fmaxf(nrm, 1e-12f);
#pragma unroll
        for (int c = 0; c < 9; ++c) Xs[i * XSTRIDE + c] = v[c] * sc;
        Xs[i * XSTRIDE + 9] = 0.0f;
    }
    if (tid < 16) Xs[N_PTS * XSTRIDE + tid] = 0.0f;   // pad tail for m>8 reads
    __syncthreads();

    // ---- Gram A = X^T X via V_WMMA_F32_16X16X4_F32, K split across 8 waves ----
    const int wave = tid >> 5;
    const int lane = tid & 31;
    const int mrow = lane & 15;      // A: M index / C: N index
    const int half = lane >> 4;      // lane group selects K pair / C row half
    v8f acc = {0.f, 0.f, 0.f, 0.f, 0.f, 0.f, 0.f, 0.f};
    const int kb0 = wave * 64;
#pragma unroll
    for (int kk = 0; kk < 64; kk += 4) {
        int k = kb0 + kk + half * 2;
        v2f a;
        a.x = Xs[k * XSTRIDE + mrow];         // A[m][k]   (== B[k][n], symmetric)
        a.y = Xs[(k + 1) * XSTRIDE + mrow];   // A[m][k+1]
        acc = __builtin_amdgcn_wmma_f32_16x16x4_f32(
            /*neg_a=*/false, a, /*neg_b=*/false, a,
            /*c_mod=*/(short)0, acc, /*reuse_a=*/false, /*reuse_b=*/false);
    }
    // C layout: VGPR v, row = v + half*8, col = lane&15
#pragma unroll
    for (int v = 0; v < 8; ++v) {
        int rr = v + half * 8;
        sPart[wave * 256 + rr * 16 + mrow] = acc[v];
    }
    __syncthreads();

    // ---- reduce 8 partial Grams, keep 9x9 ----
    {
        float sum = 0.0f;
#pragma unroll
        for (int w = 0; w < WAVES; ++w) sum += sPart[w * 256 + tid];
        int rr = tid >> 4, cc = tid & 15;
        if (rr < 9 && cc < 9) sA[rr * 9 + cc] = sum;
    }
    __syncthreads();

    // ---- eigen solves + rank-2 projection + denormalization (thread 0) ----
    if (tid == 0) {
        jacobi_sym(sA, sV, 9, 12);
        int im = 0; float mv = sA[0];
        for (int i = 1; i < 9; ++i) {
            float d = sA[i * 9 + i];
            if (d < mv) { mv = d; im = i; }
        }
        float f[9]; float nrm = 0.0f;
        for (int i = 0; i < 9; ++i) { f[i] = sV[i * 9 + im]; nrm += f[i] * f[i]; }
        nrm = sqrtf(fmaxf(nrm, 1e-30f));
        for (int i = 0; i < 9; ++i) { f[i] /= nrm; sFvec[i] = f[i]; }

        float F[3][3];
        for (int i = 0; i < 3; ++i)
            for (int j = 0; j < 3; ++j) F[i][j] = f[i * 3 + j];

        // M = F^T F, smallest eigenvector = smallest right singular vector of F
        for (int i = 0; i < 3; ++i)
            for (int j = 0; j < 3; ++j) {
                float s = 0.0f;
                for (int k = 0; k < 3; ++k) s += F[k][i] * F[k][j];
                sM3[i * 3 + j] = s;
            }
        jacobi_sym(sM3, sV3, 3, 10);
        int jm = 0; float mv3 = sM3[0];
        for (int i = 1; i < 3; ++i) {
            float d = sM3[i * 3 + i];
            if (d < mv3) { mv3 = d; jm = i; }
        }
        float v3[3]; float n3 = 0.0f;
        for (int i = 0; i < 3; ++i) { v3[i] = sV3[i * 3 + jm]; n3 += v3[i] * v3[i]; }
        n3 = sqrtf(fmaxf(n3, 1e-30f));
        for (int i = 0; i < 3; ++i) v3[i] /= n3;

        // F_rank2 = F - (F v3) v3^T   (exactly zeroes smallest singular value)
        float Fv[3];
        for (int i = 0; i < 3; ++i)
            Fv[i] = F[i][0] * v3[0] + F[i][1] * v3[1] + F[i][2] * v3[2];
        float Fr[3][3];
        for (int i = 0; i < 3; ++i)
            for (int j = 0; j < 3; ++j) Fr[i][j] = F[i][j] - Fv[i] * v3[j];

        // out = T2^T * Fr * T1
        const float T1m[3][3] = {{s1, 0.0f, -c1x * s1},
                                 {0.0f, s1, -c1y * s1},
                                 {0.0f, 0.0f, 1.0f}};
        const float T2m[3][3] = {{s2, 0.0f, -c2x * s2},
                                 {0.0f, s2, -c2y * s2},
                                 {0.0f, 0.0f, 1.0f}};
        float Tmp[3][3];
        for (int i = 0; i < 3; ++i)
            for (int j = 0; j < 3; ++j) {
                float s = 0.0f;
                for (int k = 0; k < 3; ++k) s += T2m[k][i] * Fr[k][j];
                Tmp[i][j] = s;
            }
        for (int i = 0; i < 3; ++i)
            for (int j = 0; j < 3; ++j) {
                float s = 0.0f;
                for (int k = 0; k < 3; ++k) s += Tmp[i][k] * T1m[k][j];
                outF[(size_t)b * 9 + i * 3 + j] = s;
            }
    }
    __syncthreads();

    // ---- residual = X . f_hat ----
#pragma unroll
    for (int j = 0; j < 2; ++j) {
        int i = tid + j * THREADS;
        float r = 0.0f;
#pragma unroll
        for (int c = 0; c < 9; ++c) r += Xs[i * XSTRIDE + c] * sFvec[c];
        outR[(size_t)b * N_PTS + i] = r;
    }
}

extern "C" void kernel_launch(void* const* d_in, const int* in_sizes, int n_in,
                              void* d_out, int out_size, void* d_ws, size_t ws_size,
                              hipStream_t stream) {
    const float* pts1 = (const float*)d_in[0];  // [B,N,3]
    const float* pts2 = (const float*)d_in[1];  // [B,N,3]
    const float* wts  = (const float*)d_in[2];  // [B,1,N]
    const int B = in_sizes[0] / (N_PTS * 3);

    float* outF = (float*)d_out;                 // [B,3,3] first
    float* outR = outF + (size_t)B * 9;          // [B,N] second

    fit_fundamental_kernel<<<B, THREADS, 0, stream>>>(pts1, pts2, wts, outF, outR);
}